// PredictionDecoder_77532749628078
// MI455X (gfx1250) — compile-verified
//
#include <hip/hip_runtime.h>
#include <hip/hip_bf16.h>
#include <stdint.h>

typedef __attribute__((ext_vector_type(2))) float v2f;
typedef __attribute__((ext_vector_type(4))) float v4f;
typedef __attribute__((ext_vector_type(8))) float v8f;

#define N_ANCH   8400
#define KPRE     1000
#define MAXDET   100
#define CONF_THRF 0.2f
#define IOU_THRF  0.7f
#define SORT_N   16384

// ---------------------------------------------------------------------------
// Kernel 1: DFL decode via V_WMMA_F32_16X16X4_F32.
// Each wave32 handles 4 anchors = 16 "items" (anchor-side pairs, 16 bins each).
// Lane l and l+16 both own item (l&15): they load its 16 bins (same addresses,
// coalescer merges), compute max + exp + sum, and feed the WMMA A-matrix in the
// documented 16x4 f32 layout (lanes 0-15: K=0,1 ; lanes 16-31: K=2,3).
// B holds the DFL projection r (replicated across all 16 columns), so every
// column of D is dot(exp, proj); divide by the exp-sum afterwards (exact f32).
// ---------------------------------------------------------------------------
__global__ __launch_bounds__(256) void dfl_decode_kernel(const float* __restrict__ pb,
                                                         float* __restrict__ boxes,
                                                         long long total /* B*N */) {
  const int lane = threadIdx.x & 31;
  const int hi   = lane >> 4;                       // 0: lanes 0-15, 1: lanes 16-31
  const long long wave_id = (long long)blockIdx.x * 8 + (threadIdx.x >> 5);
  const long long abase   = wave_id * 4;            // first anchor slot (b*N+n)
  if (abase >= total) return;                       // wave-uniform: EXEC stays full

  const int item = lane & 15;                       // 0..15 : (a_local<<2)|side
  const long long slot = abase + (item >> 2);
  const float* p = pb + slot * 64 + (long long)(item & 3) * 16;

  float x[16];
#pragma unroll
  for (int q = 0; q < 4; ++q) {
    v4f t = *(const v4f*)(p + q * 4);
    x[q*4+0] = t[0]; x[q*4+1] = t[1]; x[q*4+2] = t[2]; x[q*4+3] = t[3];
  }
  float mx = x[0];
#pragma unroll
  for (int r = 1; r < 16; ++r) mx = fmaxf(mx, x[r]);
  float e[16]; float s = 0.f;
#pragma unroll
  for (int r = 0; r < 16; ++r) { e[r] = __expf(x[r] - mx); s += e[r]; }

  v8f acc = {};
#pragma unroll
  for (int g = 0; g < 4; ++g) {                     // chain K=4 chunks -> K=16
    v2f a, b;
    a[0] = hi ? e[4*g+2] : e[4*g+0];                // A[m, K] per ISA 16x4 layout
    a[1] = hi ? e[4*g+3] : e[4*g+1];
    b[0] = (float)(4*g + (hi ? 2 : 0));             // B[K, n] = proj, all columns
    b[1] = (float)(4*g + (hi ? 3 : 1));
    acc = __builtin_amdgcn_wmma_f32_16x16x4_f32(false, a, false, b,
                                                (short)0, acc, false, false);
  }

  // D layout: VGPR v at lane l -> item (v + 8*hi). Normalize by that item's sum.
  float df[8];
#pragma unroll
  for (int v = 0; v < 8; ++v) {
    float sv = __shfl(s, v + (hi ? 8 : 0), 32);     // item (v+8*hi) sum lives there
    df[v] = acc[v] / sv;
  }

  // Lanes 0,1,16,17 write anchors 0..3 (each has two full anchors in df[0..7]).
  int wa = -1;
  if      (lane == 0)  wa = 0;
  else if (lane == 1)  wa = 1;
  else if (lane == 16) wa = 2;
  else if (lane == 17) wa = 3;
  if (wa >= 0) {
    long long wslot = abase + wa;
    int n = (int)(wslot % N_ANCH);
    float ax, ay, st;
    if (n < 6400)      { st = 8.f;  ax = (float)(n % 80) + 0.5f; ay = (float)(n / 80) + 0.5f; }
    else if (n < 8000) { int t = n - 6400; st = 16.f; ax = (float)(t % 40) + 0.5f; ay = (float)(t / 40) + 0.5f; }
    else               { int t = n - 8000; st = 32.f; ax = (float)(t % 20) + 0.5f; ay = (float)(t / 20) + 0.5f; }
    float d0 = (wa & 1) ? df[4] : df[0];
    float d1 = (wa & 1) ? df[5] : df[1];
    float d2 = (wa & 1) ? df[6] : df[2];
    float d3 = (wa & 1) ? df[7] : df[3];
    v4f o;
    o[0] = (ax - d0) * st; o[1] = (ay - d1) * st;
    o[2] = (ax + d2) * st; o[3] = (ay + d3) * st;
    *(v4f*)(boxes + wslot * 4) = o;
  }
}

// ---------------------------------------------------------------------------
// Kernel 2: per-anchor class max + argmax (first-max tie rule like jnp.argmax).
// ---------------------------------------------------------------------------
__global__ __launch_bounds__(256) void cls_max_kernel(const float* __restrict__ pc,
                                                      float* __restrict__ scores,
                                                      int* __restrict__ cls,
                                                      long long total) {
  long long i = (long long)blockIdx.x * blockDim.x + threadIdx.x;
  if (i >= total) return;
  const float* p = pc + i * 80;
  float best = -3.4e38f; int bi = 0;
#pragma unroll
  for (int c = 0; c < 80; c += 4) {
    v4f t = *(const v4f*)(p + c);
#pragma unroll
    for (int u = 0; u < 4; ++u)
      if (t[u] > best) { best = t[u]; bi = c + u; }
  }
  scores[i] = best;
  cls[i]    = bi;
}

// ---------------------------------------------------------------------------
// Kernel 3: per-batch descending stable sort (reproduces lax.top_k order).
// Keys: (ordered score bits << 32) | ~idx  -> desc score, asc idx on ties.
// Scores are positive (max of uniforms) so bits|0x80000000 is order-preserving;
// padding key 0 sorts last. 128 KB LDS bitonic sort of 16384 u64 keys.
// ---------------------------------------------------------------------------
__global__ __launch_bounds__(1024) void topk_sort_kernel(const float* __restrict__ scores,
                                                         float* __restrict__ top_s,
                                                         int* __restrict__ top_i) {
  __shared__ unsigned long long key[SORT_N];
  const int b = blockIdx.x;
  const float* s = scores + (long long)b * N_ANCH;
  for (int i = threadIdx.x; i < SORT_N; i += 1024) {
    unsigned long long k = 0ull;
    if (i < N_ANCH) {
      unsigned int sb = __float_as_uint(s[i]) | 0x80000000u;
      k = ((unsigned long long)sb << 32) | (unsigned int)(~i);
    }
    key[i] = k;
  }
  __syncthreads();
  for (int ksz = 2; ksz <= SORT_N; ksz <<= 1) {
    for (int j = ksz >> 1; j > 0; j >>= 1) {
      for (int t = threadIdx.x; t < SORT_N / 2; t += 1024) {
        int i   = ((t & ~(j - 1)) << 1) | (t & (j - 1));
        int ixj = i | j;
        bool desc = ((i & ksz) == 0);
        unsigned long long a = key[i], c = key[ixj];
        bool sw = desc ? (a < c) : (a > c);
        if (sw) { key[i] = c; key[ixj] = a; }
      }
      __syncthreads();
    }
  }
  for (int r = threadIdx.x; r < KPRE; r += 1024) {
    unsigned long long k = key[r];
    top_s[b * KPRE + r] = __uint_as_float((unsigned int)(k >> 32) & 0x7FFFFFFFu);
    top_i[b * KPRE + r] = (int)(~(unsigned int)(k & 0xFFFFFFFFu));
  }
}

// ---------------------------------------------------------------------------
// Kernel 4: NMS + final top-100 selection + output writes. One WG per batch.
//  - suppression bitmatrix built with one wave per 32-column block via __ballot
//  - single-wave greedy pass: keep mask in registers (1 word/lane), __shfl
//    broadcast of the decision bit; identical semantics to reference fori_loop
//  - prefix-sum compaction reproduces top_k(masked,100): valid entries in
//    score order, then invalid entries (value -1) in index order.
// ---------------------------------------------------------------------------
__global__ __launch_bounds__(1024) void nms_out_kernel(const float* __restrict__ boxes,
                                                       const int* __restrict__ cls,
                                                       const float* __restrict__ dists,
                                                       const float* __restrict__ top_s,
                                                       const int* __restrict__ top_i,
                                                       float* __restrict__ out,
                                                       int B) {
  __shared__ float bx1[KPRE], by1[KPRE], bx2[KPRE], by2[KPRE], bar[KPRE];
  __shared__ unsigned int supr[KPRE * 32];   // 1000 rows x 1024 bits
  __shared__ unsigned int keepw[32];
  __shared__ unsigned char svalid[KPRE];
  __shared__ int scan[1024];
  __shared__ int sel[MAXDET];

  const int b    = blockIdx.x;
  const int tid  = threadIdx.x;
  const int wv   = tid >> 5;
  const int lane = tid & 31;

  // stage top-1000 boxes into LDS
  for (int i = tid; i < KPRE; i += 1024) {
    int oi = top_i[b * KPRE + i];
    const float* bp = boxes + ((long long)b * N_ANCH + oi) * 4;
    float x1 = bp[0], y1 = bp[1], x2 = bp[2], y2 = bp[3];
    bx1[i] = x1; by1[i] = y1; bx2[i] = x2; by2[i] = y2;
    bar[i] = (x2 - x1) * (y2 - y1);
  }
  __syncthreads();

  // suppression matrix: wave wv owns column block wv, lane -> column j
  {
    int j = wv * 32 + lane;
    bool jv = j < KPRE;
    int jc = jv ? j : 0;
    float jx1 = bx1[jc], jy1 = by1[jc], jx2 = bx2[jc], jy2 = by2[jc], jar = bar[jc];
    for (int i = 0; i < KPRE; ++i) {
      float lx = fmaxf(bx1[i], jx1), ly = fmaxf(by1[i], jy1);
      float rx = fminf(bx2[i], jx2), ry = fminf(by2[i], jy2);
      float iw = fmaxf(rx - lx, 0.f), ih = fmaxf(ry - ly, 0.f);
      float inter = iw * ih;
      float iou = inter / (bar[i] + jar - inter + 1e-7f);
      bool sup = jv && (j > i) && (iou > IOU_THRF);
      unsigned int word = (unsigned int)__ballot(sup);
      if (lane == 0) supr[i * 32 + wv] = word;
    }
  }
  __syncthreads();

  // greedy suppression: wave 0 only, keep mask in registers
  if (tid < 32) {
    unsigned int keep = 0xFFFFFFFFu;
    for (int i = 0; i < KPRE; ++i) {
      unsigned int kw  = (unsigned int)__shfl((int)keep, i >> 5, 32);
      unsigned int row = supr[i * 32 + lane];
      unsigned int en  = 0u - ((kw >> (i & 31)) & 1u);   // all-ones iff keep[i]
      keep &= ~(row & en);
    }
    keepw[lane] = keep;
  }
  __syncthreads();

  for (int i = tid; i < KPRE; i += 1024) {
    bool v = ((keepw[i >> 5] >> (i & 31)) & 1u) && (top_s[b * KPRE + i] > CONF_THRF);
    svalid[i] = v ? 1 : 0;
  }
  __syncthreads();

  // inclusive prefix sum of valid flags (Hillis-Steele, 1024 threads)
  scan[tid] = (tid < KPRE) ? (int)svalid[tid] : 0;
  __syncthreads();
  for (int off = 1; off < 1024; off <<= 1) {
    int v = scan[tid];
    int u = (tid >= off) ? scan[tid - off] : 0;
    __syncthreads();
    scan[tid] = v + u;
    __syncthreads();
  }
  int nvalid = scan[1023];

  if (tid < KPRE) {
    int inc  = scan[tid];
    int vfl  = (int)svalid[tid];
    int excl = inc - vfl;                  // valid entries before tid
    if (vfl && excl < MAXDET) sel[excl] = tid;
    int pos2 = nvalid + (tid - inc);       // invalid entries before tid, after valids
    if (!vfl && pos2 < MAXDET) sel[pos2] = tid;
  }
  __syncthreads();

  // write outputs: boxes | scores | cls | dist | valid (all as f32)
  float* obox   = out;
  float* osc    = obox  + (long long)B * MAXDET * 4;
  float* ocls   = osc   + (long long)B * MAXDET;
  float* odist  = ocls  + (long long)B * MAXDET;
  float* ovalid = odist + (long long)B * MAXDET;
  for (int t = tid; t < MAXDET; t += 1024) {
    int i  = sel[t];
    bool v = svalid[i] != 0;
    int oi = top_i[b * KPRE + i];
    long long slot = (long long)b * MAXDET + t;
    obox[slot * 4 + 0] = bx1[i];
    obox[slot * 4 + 1] = by1[i];
    obox[slot * 4 + 2] = bx2[i];
    obox[slot * 4 + 3] = by2[i];
    osc[slot]    = v ? top_s[b * KPRE + i] : -1.0f;
    ocls[slot]   = (float)cls[(long long)b * N_ANCH + oi];
    odist[slot]  = dists[(long long)b * N_ANCH + oi];
    ovalid[slot] = v ? 1.0f : 0.0f;
  }
}

// ---------------------------------------------------------------------------
extern "C" void kernel_launch(void* const* d_in, const int* in_sizes, int n_in,
                              void* d_out, int out_size, void* d_ws, size_t ws_size,
                              hipStream_t stream) {
  const float* pb = (const float*)d_in[0];   // (B, N, 64)
  const float* pc = (const float*)d_in[1];   // (B, N, 80)
  const float* pd = (const float*)d_in[2];   // (B, N, 1)
  // d_in[3] images: unused by the computation

  const long long total = (long long)in_sizes[0] / 64;   // B*N
  const int B = (int)(total / N_ANCH);

  // workspace carve-out (all 16B aligned)
  char* w = (char*)d_ws;
  float* ws_boxes  = (float*)w; w += (size_t)total * 4 * sizeof(float);
  float* ws_scores = (float*)w; w += (size_t)total * sizeof(float);
  int*   ws_cls    = (int*)w;   w += (size_t)total * sizeof(int);
  float* ws_tops   = (float*)w; w += (size_t)B * KPRE * sizeof(float);
  int*   ws_topi   = (int*)w;

  // K1: DFL + box decode (WMMA). 8 waves/block * 4 anchors/wave = 32 anchors.
  int blocks1 = (int)((total + 31) / 32);
  dfl_decode_kernel<<<blocks1, 256, 0, stream>>>(pb, ws_boxes, total);

  // K2: class max / argmax
  int blocks2 = (int)((total + 255) / 256);
  cls_max_kernel<<<blocks2, 256, 0, stream>>>(pc, ws_scores, ws_cls, total);

  // K3: per-batch stable descending sort -> top-1000
  topk_sort_kernel<<<B, 1024, 0, stream>>>(ws_scores, ws_tops, ws_topi);

  // K4: NMS + selection + outputs
  nms_out_kernel<<<B, 1024, 0, stream>>>(ws_boxes, ws_cls, pd, ws_tops, ws_topi,
                                         (float*)d_out, B);
}